// InteractionBlock_49117245997498
// MI455X (gfx1250) — compile-verified
//
#include <hip/hip_runtime.h>
#include <cstdint>
#include <cstddef>

// CDNA5 / gfx1250. fp32 end-to-end via V_WMMA_F32_16X16X4_F32 (workload is
// HBM-bound; fp32 matrix throughput is nowhere near the bottleneck).
// Round 2: aggressive fusion to cut HBM round-trips (rbf_e, x_kj, t1 of every
// residual block never touch global memory).

typedef float v2f __attribute__((ext_vector_type(2)));
typedef float v8f __attribute__((ext_vector_type(8)));

#define HID 128
#define INTC 64
#define SBF_W 42
#define RBF_W 6

__device__ __forceinline__ float silu_f(float v) {
    return v * (1.0f / (1.0f + __expf(-v)));
}

// ---------------------------------------------------------------------------
// WMMA 16x16x4 f32 tile: acc(16x16) += lA(16 rows x K) @ lW(K x cols@c0)
// VGPR layout (ISA 7.12.2): A/B: V0 <- {k0 | k0+2}, V1 <- {k0+1 | k0+3}
// (half-wave split); C/D: V_r <- row r | r+8, col = lane%16.
// ---------------------------------------------------------------------------
template <int AP, int WP, int K>
__device__ __forceinline__ v8f wmma_tile(const float* __restrict__ lA,
                                         const float* __restrict__ lW,
                                         int r0, int c0, int lrow, int hi)
{
    v8f acc = {};
    #pragma unroll 8
    for (int s = 0; s < K / 4; ++s) {
        const int ka = s * 4 + hi * 2;
        float2 av = *reinterpret_cast<const float2*>(&lA[(r0 + lrow) * AP + ka]);
        v2f a, b;
        a.x = av.x;
        a.y = av.y;
        b.x = lW[ka * WP + c0 + lrow];
        b.y = lW[(ka + 1) * WP + c0 + lrow];
        acc = __builtin_amdgcn_wmma_f32_16x16x4_f32(
            false, a, false, b, (short)0, acc, false, false);
    }
    return acc;
}

// Cooperative stagers (256 threads, float4 moves, padded LDS strides).
__device__ __forceinline__ void stage_rows(const float* __restrict__ g,
                                           float* __restrict__ l,
                                           int rowBase, int M, int K, int P, int tid)
{
    for (int c = tid; c < (128 * K) / 4; c += 256) {
        int lin = c * 4;
        int r = lin / K, k = lin % K;
        float4 v = make_float4(0.f, 0.f, 0.f, 0.f);
        if (rowBase + r < M)
            v = *reinterpret_cast<const float4*>(&g[(size_t)(rowBase + r) * K + k]);
        *reinterpret_cast<float4*>(&l[r * P + k]) = v;
    }
}

__device__ __forceinline__ void stage_w(const float* __restrict__ g,
                                        float* __restrict__ l,
                                        int K, int N, int P, int tid)
{
    for (int c = tid; c < (K * N) / 4; c += 256) {
        int lin = c * 4;
        int k = lin / N, n = lin % N;
        float4 v = *reinterpret_cast<const float4*>(&g[(size_t)k * N + n]);
        *reinterpret_cast<float4*>(&l[k * P + n]) = v;
    }
}

// ---------------------------------------------------------------------------
// Fuse tiny two-layer basis embeddings:
//   Wrbf [6,128] = rbf1[6,8] @ rbf2[8,128];  Wsbf [42,64] = sbf1 @ sbf2
// ---------------------------------------------------------------------------
__global__ __launch_bounds__(256) void fuse_weights_kernel(
    const float* __restrict__ rbf1, const float* __restrict__ rbf2,
    const float* __restrict__ sbf1, const float* __restrict__ sbf2,
    float* __restrict__ Wrbf, float* __restrict__ Wsbf)
{
    int tid = blockIdx.x * 256 + threadIdx.x;
    int stride = gridDim.x * 256;
    for (int i = tid; i < RBF_W * HID; i += stride) {
        int k = i >> 7, n = i & 127;
        float s = 0.f;
        #pragma unroll
        for (int b = 0; b < 8; ++b) s += rbf1[k * 8 + b] * rbf2[b * HID + n];
        Wrbf[i] = s;
    }
    for (int i = tid; i < SBF_W * INTC; i += stride) {
        int k = i >> 6, n = i & 63;
        float s = 0.f;
        #pragma unroll
        for (int b = 0; b < 8; ++b) s += sbf1[k * 8 + b] * sbf2[b * INTC + n];
        Wsbf[i] = s;
    }
}

// ---------------------------------------------------------------------------
// FRONT FUSION. Per 128-row block of x:
//   x_ji   = silu(x @ Wji + bji)                       -> global
//   x_kj   = silu(x @ Wkj + bkj) * (rbf @ Wrbf)        -> LDS only
//   xkjd   = silu(x_kj @ Wdown)                        -> global [.,64]
// LDS: lA(x) + lW(weights, reused) + lB(x_kj) + Wrbf + rbf rows  (~210 KB)
// ---------------------------------------------------------------------------
__global__ __launch_bounds__(256) void front_kernel(
    const float* __restrict__ x, const float* __restrict__ rbf,
    const float* __restrict__ Wji, const float* __restrict__ bji,
    const float* __restrict__ Wkj, const float* __restrict__ bkj,
    const float* __restrict__ Wrbf, const float* __restrict__ Wdown,
    float* __restrict__ x_ji, float* __restrict__ xkjd, int M)
{
    constexpr int AP = HID + 4;              // 132
    extern __shared__ float smem[];
    float* lA    = smem;                     // 128 x 132   (x block)
    float* lW    = lA + 128 * AP;            // 128 x 132   (Wji -> Wkj -> Wdown)
    float* lB    = lW + 128 * AP;            // 128 x 132   (x_kj block)
    float* lWrbf = lB + 128 * AP;            // 6 x 128
    float* lrbf  = lWrbf + RBF_W * HID;      // 128 x 8 (stride 8, 6 used)

    const int tid = threadIdx.x;
    const int rowBase = blockIdx.x * 128;

    stage_rows(x, lA, rowBase, M, HID, AP, tid);
    stage_w(Wji, lW, HID, HID, AP, tid);
    for (int i = tid; i < RBF_W * HID; i += 256) lWrbf[i] = Wrbf[i];
    for (int i = tid; i < 128 * RBF_W; i += 256) {
        int r = i / RBF_W, k = i % RBF_W;
        lrbf[r * 8 + k] = (rowBase + r < M) ? rbf[(size_t)(rowBase + r) * RBF_W + k] : 0.f;
    }
    __syncthreads();

    const int wv = tid >> 5, lane = tid & 31;
    const int hi = (lane >= 16) ? 1 : 0;
    const int lrow = lane & 15;
    const int r0 = wv * 16;
    const int rb = hi * 8;

    // phase 1: x_ji
    for (int t = 0; t < 8; ++t) {
        v8f acc = wmma_tile<AP, AP, HID>(lA, lW, r0, t * 16, lrow, hi);
        const int col = t * 16 + lrow;
        const float bb = bji[col];
        #pragma unroll
        for (int r = 0; r < 8; ++r) {
            const int grow = rowBase + r0 + rb + r;
            if (grow < M) x_ji[(size_t)grow * HID + col] = silu_f(acc[r] + bb);
        }
    }
    __syncthreads();
    stage_w(Wkj, lW, HID, HID, AP, tid);
    __syncthreads();

    // phase 2: x_kj = silu(x@Wkj+b) * rbf_e  -> lB
    for (int t = 0; t < 8; ++t) {
        v8f acc = wmma_tile<AP, AP, HID>(lA, lW, r0, t * 16, lrow, hi);
        const int col = t * 16 + lrow;
        const float bb = bkj[col];
        #pragma unroll
        for (int r = 0; r < 8; ++r) {
            const int lr = r0 + rb + r;
            float g = 0.f;
            #pragma unroll
            for (int k = 0; k < RBF_W; ++k) g += lrbf[lr * 8 + k] * lWrbf[k * HID + col];
            lB[lr * AP + col] = silu_f(acc[r] + bb) * g;
        }
    }
    __syncthreads();
    stage_w(Wdown, lW, HID, INTC, INTC + 4, tid);
    __syncthreads();

    // phase 3: xkjd = silu(x_kj @ Wdown)   [.,64]
    for (int t = 0; t < 4; ++t) {
        v8f acc = wmma_tile<AP, INTC + 4, HID>(lB, lW, r0, t * 16, lrow, hi);
        const int col = t * 16 + lrow;
        #pragma unroll
        for (int r = 0; r < 8; ++r) {
            const int grow = rowBase + r0 + rb + r;
            if (grow < M) xkjd[(size_t)grow * INTC + col] = silu_f(acc[r]);
        }
    }
}

// ---------------------------------------------------------------------------
// Generic WMMA GEMM with fused epilogue: (+bias) -> SiLU -> (*mul) -> (+add)
// ---------------------------------------------------------------------------
template <int K, int N>
__global__ __launch_bounds__(256) void gemm_wmma(
    const float* __restrict__ A, const float* __restrict__ W,
    const float* __restrict__ bias, const float* __restrict__ mul_aux,
    const float* __restrict__ add_aux, float* __restrict__ out,
    int M, int act)
{
    constexpr int AP = K + 4;
    constexpr int WP = N + 4;
    extern __shared__ float smem[];
    float* lA = smem;
    float* lW = smem + 128 * AP;

    const int tid = threadIdx.x;
    const int rowBase = blockIdx.x * 128;

    stage_rows(A, lA, rowBase, M, K, AP, tid);
    stage_w(W, lW, K, N, WP, tid);
    __syncthreads();

    const int wv = tid >> 5, lane = tid & 31;
    const int hi = (lane >= 16) ? 1 : 0;
    const int lrow = lane & 15;
    const int r0 = wv * 16;
    const int rb = hi * 8;

    for (int t = 0; t < N / 16; ++t) {
        v8f acc = wmma_tile<AP, WP, K>(lA, lW, r0, t * 16, lrow, hi);
        const int col = t * 16 + lrow;
        #pragma unroll
        for (int r = 0; r < 8; ++r) {
            const int grow = rowBase + r0 + rb + r;
            if (grow < M) {
                float v = acc[r];
                if (bias)    v += bias[col];
                if (act)     v = silu_f(v);
                if (mul_aux) v *= mul_aux[(size_t)grow * N + col];
                if (add_aux) v += add_aux[(size_t)grow * N + col];
                out[(size_t)grow * N + col] = v;
            }
        }
    }
}

// ---------------------------------------------------------------------------
// FUSED RESIDUAL BLOCK:  out = h + silu(W2 @ silu(W1 @ h + b1) + b2)
// N == K == 128, so a 128-row block computes complete rows of t1: each wave
// keeps its 16x128 t1 in 64 VGPRs, rewrites it into lA, swaps W1->W2 in lW,
// and runs the second GEMM. t1 never touches global memory.
// ---------------------------------------------------------------------------
__global__ __launch_bounds__(256) void residual_fused(
    const float* __restrict__ h,
    const float* __restrict__ W1, const float* __restrict__ b1,
    const float* __restrict__ W2, const float* __restrict__ b2,
    float* __restrict__ out, int M)
{
    constexpr int AP = HID + 4;
    extern __shared__ float smem[];
    float* lA = smem;            // h block -> t1 block
    float* lW = smem + 128 * AP; // W1 -> W2

    const int tid = threadIdx.x;
    const int rowBase = blockIdx.x * 128;

    stage_rows(h, lA, rowBase, M, HID, AP, tid);
    stage_w(W1, lW, HID, HID, AP, tid);
    __syncthreads();

    const int wv = tid >> 5, lane = tid & 31;
    const int hi = (lane >= 16) ? 1 : 0;
    const int lrow = lane & 15;
    const int r0 = wv * 16;
    const int rb = hi * 8;

    // phase 1: t1 = silu(h @ W1 + b1), kept in registers (8 tiles x v8f)
    v8f t1v[8];
    #pragma unroll
    for (int t = 0; t < 8; ++t) {
        v8f acc = wmma_tile<AP, AP, HID>(lA, lW, r0, t * 16, lrow, hi);
        const float bb = b1[t * 16 + lrow];
        #pragma unroll
        for (int r = 0; r < 8; ++r) acc[r] = silu_f(acc[r] + bb);
        t1v[t] = acc;
    }
    __syncthreads();   // all waves done reading lA (h) and lW (W1)

    // write t1 into lA (own rows only); swap in W2
    #pragma unroll
    for (int t = 0; t < 8; ++t) {
        #pragma unroll
        for (int r = 0; r < 8; ++r)
            lA[(r0 + rb + r) * AP + t * 16 + lrow] = t1v[t][r];
    }
    stage_w(W2, lW, HID, HID, AP, tid);
    __syncthreads();

    // phase 2: out = h + silu(t1 @ W2 + b2)   (h re-read is L2-hot)
    for (int t = 0; t < 8; ++t) {
        v8f acc = wmma_tile<AP, AP, HID>(lA, lW, r0, t * 16, lrow, hi);
        const int col = t * 16 + lrow;
        const float bb = b2[col];
        #pragma unroll
        for (int r = 0; r < 8; ++r) {
            const int grow = rowBase + r0 + rb + r;
            if (grow < M) {
                float v = silu_f(acc[r] + bb) + h[(size_t)grow * HID + col];
                out[(size_t)grow * HID + col] = v;
            }
        }
    }
}

// ---------------------------------------------------------------------------
// Triplet phase: sbf_e = sbf[t]@Wsbf (LDS weight); gather xkjd[idx_kj];
// scatter-add into agg[idx_ji] with relaxed fp32 atomics (L2-resident).
// ---------------------------------------------------------------------------
__global__ __launch_bounds__(256) void triplet_kernel(
    const float* __restrict__ sbf, const float* __restrict__ Wsbf,
    const float* __restrict__ xkjd,
    const int* __restrict__ idx_kj, const int* __restrict__ idx_ji,
    float* __restrict__ agg, int T)
{
    __shared__ float lW[SBF_W * INTC];
    __shared__ float lsbf[4][SBF_W + 2];
    for (int i = threadIdx.x; i < SBF_W * INTC; i += 256) lW[i] = Wsbf[i];

    const int rgrp = threadIdx.x >> 6;
    const int col = threadIdx.x & 63;
    const int row = blockIdx.x * 4 + rgrp;
    if (row < T && col < SBF_W)
        lsbf[rgrp][col] = sbf[(size_t)row * SBF_W + col];
    __syncthreads();

    if (row < T) {
        float s = 0.f;
        #pragma unroll
        for (int k = 0; k < SBF_W; ++k) s += lsbf[rgrp][k] * lW[k * INTC + col];
        const int kj = idx_kj[row];
        const int ji = idx_ji[row];
        const float m = s * xkjd[(size_t)kj * INTC + col];
        __hip_atomic_fetch_add(&agg[(size_t)ji * INTC + col], m,
                               __ATOMIC_RELAXED, __HIP_MEMORY_SCOPE_AGENT);
    }
}

// ---------------------------------------------------------------------------
// Host orchestration. Input order (JAX pytree, dict keys sorted):
// x, rbf, sbf, idx_kj, idx_ji, down.W, ji.W, ji.b, kj.W, kj.b, lin.W, lin.b,
// rbf1.W, rbf2.W, res_after[0].{l1.W,l1.b,l2.W,l2.b}, res_after[1].{...},
// res_before[0].{...}, sbf1.W, sbf2.W, up.W
// ---------------------------------------------------------------------------
extern "C" void kernel_launch(void* const* d_in, const int* in_sizes, int n_in,
                              void* d_out, int out_size, void* d_ws, size_t ws_size,
                              hipStream_t stream)
{
    const float* x      = (const float*)d_in[0];
    const float* rbf    = (const float*)d_in[1];
    const float* sbf    = (const float*)d_in[2];
    const int*   idx_kj = (const int*)d_in[3];
    const int*   idx_ji = (const int*)d_in[4];

    const float* W_down = (const float*)d_in[5];
    const float* W_ji   = (const float*)d_in[6];
    const float* b_ji   = (const float*)d_in[7];
    const float* W_kj   = (const float*)d_in[8];
    const float* b_kj   = (const float*)d_in[9];
    const float* W_lin  = (const float*)d_in[10];
    const float* b_lin  = (const float*)d_in[11];
    const float* W_rbf1 = (const float*)d_in[12];
    const float* W_rbf2 = (const float*)d_in[13];
    const float* Wa0l1  = (const float*)d_in[14];
    const float* ba0l1  = (const float*)d_in[15];
    const float* Wa0l2  = (const float*)d_in[16];
    const float* ba0l2  = (const float*)d_in[17];
    const float* Wa1l1  = (const float*)d_in[18];
    const float* ba1l1  = (const float*)d_in[19];
    const float* Wa1l2  = (const float*)d_in[20];
    const float* ba1l2  = (const float*)d_in[21];
    const float* Wb0l1  = (const float*)d_in[22];
    const float* bb0l1  = (const float*)d_in[23];
    const float* Wb0l2  = (const float*)d_in[24];
    const float* bb0l2  = (const float*)d_in[25];
    const float* W_sbf1 = (const float*)d_in[26];
    const float* W_sbf2 = (const float*)d_in[27];
    const float* W_up   = (const float*)d_in[28];

    const int E = in_sizes[0] / HID;
    const int T = in_sizes[3];

    // Workspace layout (floats): 2 big [E,128] ping-pong + 2 [E,64].
    float* ws   = (float*)d_ws;
    float* wRbf = ws;                          // 6*128
    float* wSbf = ws + 1024;                   // 42*64
    float* bufA = ws + 4096;                   // [E,128] x_ji -> h ping
    float* bufB = bufA + (size_t)E * HID;      // [E,128] h pong
    float* bufD = bufB + (size_t)E * HID;      // [E,64]  x_kj_down
    float* bufG = bufD + (size_t)E * INTC;     // [E,64]  agg
    float* out  = (float*)d_out;               // [E,128]

    constexpr int AP = HID + 4;
    const int sh_front   = (3 * 128 * AP + RBF_W * HID + 128 * 8) * 4;    // ~210 KB
    const int sh_res     = (2 * 128 * AP) * 4;                            // 135168
    const int sh_128_128 = (128 * AP + 128 * (HID + 4)) * 4;              // 135168
    const int sh_64_128  = (128 * (64 + 4) + 64 * (HID + 4)) * 4;         //  68608
    hipFuncSetAttribute((const void*)&front_kernel,
                        hipFuncAttributeMaxDynamicSharedMemorySize, sh_front);
    hipFuncSetAttribute((const void*)&residual_fused,
                        hipFuncAttributeMaxDynamicSharedMemorySize, sh_res);
    hipFuncSetAttribute((const void*)&gemm_wmma<128, 128>,
                        hipFuncAttributeMaxDynamicSharedMemorySize, sh_128_128);
    hipFuncSetAttribute((const void*)&gemm_wmma<64, 128>,
                        hipFuncAttributeMaxDynamicSharedMemorySize, sh_64_128);

    const int gE = (E + 127) / 128;
    dim3 blk(256);

    // 1) fuse tiny embedding weights
    fuse_weights_kernel<<<11, blk, 0, stream>>>(W_rbf1, W_rbf2, W_sbf1, W_sbf2, wRbf, wSbf);
    // 2) front fusion: x_ji -> bufA, x_kj_down -> bufD (rbf_e, x_kj stay in LDS)
    front_kernel<<<gE, blk, sh_front, stream>>>(x, rbf, W_ji, b_ji, W_kj, b_kj,
                                                wRbf, W_down, bufA, bufD, E);
    // 3) agg = 0; triplet gather*embed -> scatter-add          -> bufG
    hipMemsetAsync(bufG, 0, (size_t)E * INTC * sizeof(float), stream);
    triplet_kernel<<<(T + 3) / 4, blk, 0, stream>>>(sbf, wSbf, bufD, idx_kj, idx_ji, bufG, T);
    // 4) h = x_ji + agg @ W_up                                 -> bufB
    gemm_wmma<64, 128><<<gE, blk, sh_64_128, stream>>>(bufG, W_up, nullptr, nullptr, bufA, bufB, E, 0);
    // 5) res_before[0] (fused pair)                            bufB -> bufA
    residual_fused<<<gE, blk, sh_res, stream>>>(bufB, Wb0l1, bb0l1, Wb0l2, bb0l2, bufA, E);
    // 6) h = silu(h @ W_lin + b) + x                           bufA -> bufB
    gemm_wmma<128, 128><<<gE, blk, sh_128_128, stream>>>(bufA, W_lin, b_lin, nullptr, x, bufB, E, 1);
    // 7) res_after[0] (fused)                                  bufB -> bufA
    residual_fused<<<gE, blk, sh_res, stream>>>(bufB, Wa0l1, ba0l1, Wa0l2, ba0l2, bufA, E);
    // 8) res_after[1] (fused)                                  bufA -> d_out
    residual_fused<<<gE, blk, sh_res, stream>>>(bufA, Wa1l1, ba1l1, Wa1l2, ba1l2, out, E);
}